// GraphDensenet_19937238188614
// MI455X (gfx1250) — compile-verified
//
#include <hip/hip_runtime.h>
#include <math.h>

typedef __attribute__((ext_vector_type(2))) float v2f;
typedef __attribute__((ext_vector_type(8))) float v8f;

#define GROW 12
#define CMAXP 160   // padded max channel count for LDS staging

// ---------------------------------------------------------------------------
// Weight transpose: W[K,Nc] (row-major) -> Wt[Nc,K] (row-major). Tiny, one-shot.
// ---------------------------------------------------------------------------
__global__ void transpose_w(const float* __restrict__ W, int K, int Nc,
                            float* __restrict__ Wt)
{
    const int t = blockIdx.x * blockDim.x + threadIdx.x;
    if (t >= K * Nc) return;
    const int k = t / Nc, n = t - k * Nc;
    Wt[(size_t)n * K + k] = W[t];
}

// ---------------------------------------------------------------------------
// fp32 WMMA GEMM: C[M,Nc] = act(A[M,K] @ W[K,Nc] + bias), LeakyReLU `slope`
// (slope=1 -> identity). W supplied PRE-TRANSPOSED as Wt[Nc,K] so both A and
// B fragments are contiguous b64 loads with immediate offsets. One wave per
// 16x32 output tile (two WMMA accumulators sharing the A fragment).
// REQUIRES K % 4 == 0 (true for all GEMMs here). Edge tiles handled by index
// clamping: an OOB A row / B column only feeds D rows/cols never stored, so
// loads are unconditional and EXEC stays all-1s through the WMMA region.
// ---------------------------------------------------------------------------
__global__ __launch_bounds__(32)
void wmma_gemm_bias_act(const float* __restrict__ A, int M, int K, int lda,
                        const float* __restrict__ Wt, int Nc,
                        const float* __restrict__ bias,
                        float* __restrict__ C, int ldc, float slope)
{
    const int lane = threadIdx.x;      // 0..31
    const int half = lane >> 4;        // 0 or 1
    const int l15  = lane & 15;
    const int row  = blockIdx.y * 16 + l15;    // A row this lane feeds
    const int col0 = blockIdx.x * 32 + l15;    // first B col this lane feeds
    const int col1 = col0 + 16;                // second B col

    const int rowc  = min(row,  M - 1);        // clamped (safe-garbage) indices
    const int col0c = min(col0, Nc - 1);
    const int col1c = min(col1, Nc - 1);

    // lane l feeds K-slots {half*2, half*2+1} of each 4-wide K step
    const float* __restrict__ Ar  = A  + (size_t)rowc  * lda + half * 2;
    const float* __restrict__ B0r = Wt + (size_t)col0c * K   + half * 2;
    const float* __restrict__ B1r = Wt + (size_t)col1c * K   + half * 2;

    v8f acc0 = {}, acc1 = {};
#pragma unroll 4
    for (int k0 = 0; k0 < K; k0 += 4) {
        v2f a  = *(const v2f*)(Ar  + k0);   // A[row, k0+half*2 .. +1]
        v2f b0 = *(const v2f*)(B0r + k0);   // W[k0+half*2 .. +1, col0]
        v2f b1 = *(const v2f*)(B1r + k0);   // W[k0+half*2 .. +1, col1]
        acc0 = __builtin_amdgcn_wmma_f32_16x16x4_f32(
            false, a, false, b0, (short)0, acc0, false, false);
        acc1 = __builtin_amdgcn_wmma_f32_16x16x4_f32(
            false, a, false, b1, (short)0, acc1, false, false);
    }

    // C/D layout: lane l, vgpr r -> (row = r + (l>>4)*8, col = l&15)
    const bool c0ok = col0 < Nc;
    const bool c1ok = col1 < Nc;
    const float bv0 = (bias && c0ok) ? bias[col0] : 0.f;
    const float bv1 = (bias && c1ok) ? bias[col1] : 0.f;
#pragma unroll
    for (int r = 0; r < 8; ++r) {
        const int orow = blockIdx.y * 16 + r + half * 8;
        if (orow < M) {
            if (c0ok) {
                float v = acc0[r] + bv0;
                C[(size_t)orow * ldc + col0] = (v >= 0.f) ? v : slope * v;
            }
            if (c1ok) {
                float v = acc1[r] + bv1;
                C[(size_t)orow * ldc + col1] = (v >= 0.f) ? v : slope * v;
            }
        }
    }
}

// ---------------------------------------------------------------------------
// BN statistics: one block per channel, block-reduce sum / sumsq over N rows.
// ---------------------------------------------------------------------------
__global__ void bn_stats(const float* __restrict__ X, int N, int ld,
                         float* __restrict__ sums, float* __restrict__ sqs)
{
    const int ch = blockIdx.x;
    float s = 0.f, q = 0.f;
    for (int i = threadIdx.x; i < N; i += blockDim.x) {
        float v = X[(size_t)i * ld + ch];
        s += v; q += v * v;
    }
    __shared__ float ss[256], qq[256];
    ss[threadIdx.x] = s; qq[threadIdx.x] = q;
    __syncthreads();
    for (int off = blockDim.x >> 1; off > 0; off >>= 1) {
        if ((int)threadIdx.x < off) {
            ss[threadIdx.x] += ss[threadIdx.x + off];
            qq[threadIdx.x] += qq[threadIdx.x + off];
        }
        __syncthreads();
    }
    if (threadIdx.x == 0) { sums[ch] = ss[0]; sqs[ch] = qq[0]; }
}

// ---------------------------------------------------------------------------
// Dense-block layer: xn = LeakyReLU(BN(x[:, :c])); append xn @ Wc^T (12 cols).
// Scale/shift and the 12xc weight tile staged in LDS; one thread per node.
// ---------------------------------------------------------------------------
__global__ void bn_act_conv_append(const float* __restrict__ X, int N, int ld, int c,
                                   const float* __restrict__ Wc, int wpitch,
                                   const float* __restrict__ gamma,
                                   const float* __restrict__ beta,
                                   const float* __restrict__ sums,
                                   const float* __restrict__ sqs,
                                   float* __restrict__ Xout)
{
    __shared__ float sScale[CMAXP], sShift[CMAXP], sW[GROW * CMAXP];
    for (int j = threadIdx.x; j < c; j += blockDim.x) {
        float mu  = sums[j] / (float)N;
        float var = sqs[j] / (float)N - mu * mu;
        float inv = rsqrtf(var + 1e-5f);
        float sc  = gamma[j] * inv;
        sScale[j] = sc;
        sShift[j] = beta[j] - mu * sc;
    }
    for (int t = threadIdx.x; t < GROW * c; t += blockDim.x) {
        int g = t / c, j = t - g * c;
        sW[g * CMAXP + j] = Wc[(size_t)g * wpitch + j];
    }
    __syncthreads();

    const int i = blockIdx.x * blockDim.x + threadIdx.x;
    if (i >= N) return;
    float acc[GROW];
#pragma unroll
    for (int g = 0; g < GROW; ++g) acc[g] = 0.f;
    const float* xr = X + (size_t)i * ld;
    for (int j = 0; j < c; ++j) {
        float v = xr[j] * sScale[j] + sShift[j];
        v = (v >= 0.f) ? v : 0.01f * v;
#pragma unroll
        for (int g = 0; g < GROW; ++g) acc[g] += v * sW[g * CMAXP + j];
    }
    float* xo = Xout + (size_t)i * ld + c;
#pragma unroll
    for (int g = 0; g < GROW; ++g) xo[g] = acc[g];
}

// ---------------------------------------------------------------------------
// GAT helpers
// ---------------------------------------------------------------------------
__global__ void fill_f32(float* __restrict__ p, size_t n, float v)
{
    size_t i = (size_t)blockIdx.x * blockDim.x + threadIdx.x;
    if (i < n) p[i] = v;
}

__global__ void gat_node_proj(const float* __restrict__ H, int N, int F,
                              const float* __restrict__ a,
                              float* __restrict__ pdst, float* __restrict__ psrc)
{
    const int i = blockIdx.x * blockDim.x + threadIdx.x;
    if (i >= N) return;
    const float* h = H + (size_t)i * F;
    float d = 0.f, s = 0.f;
    for (int f = 0; f < F; ++f) { float v = h[f]; d += v * a[f]; s += v * a[F + f]; }
    pdst[i] = d; psrc[i] = s;
}

__device__ inline void atomicMaxFloat(float* addr, float v)
{
    // sign-split trick: lowers to global_atomic_max_i32 / global_atomic_min_u32
    if (v >= 0.f) atomicMax((int*)addr, __float_as_int(v));
    else          atomicMin((unsigned int*)addr, __float_as_uint(v));
}

__global__ void edge_logit_max(const int* __restrict__ src, const int* __restrict__ dst,
                               int E, const float* __restrict__ pdst,
                               const float* __restrict__ psrc,
                               float* __restrict__ e, float* __restrict__ m)
{
    const int t = blockIdx.x * blockDim.x + threadIdx.x;
    if (t >= E) return;
    const int d = dst[t];
    float v = pdst[d] + psrc[src[t]];
    v = (v >= 0.f) ? v : 0.2f * v;   // GAT attention slope
    e[t] = v;
    atomicMaxFloat(&m[d], v);
}

__global__ void fix_m(float* __restrict__ m, int N)
{
    const int i = blockIdx.x * blockDim.x + threadIdx.x;
    if (i >= N) return;
    float v = m[i];
    if (!(v > -3.0e38f)) m[i] = 0.f;   // -inf (no incoming edges) -> 0
}

__global__ void edge_exp_den(const int* __restrict__ dst, int E,
                             const float* __restrict__ e, const float* __restrict__ m,
                             float* __restrict__ ex, float* __restrict__ den)
{
    const int t = blockIdx.x * blockDim.x + threadIdx.x;
    if (t >= E) return;
    const int d = dst[t];
    float v = __expf(e[t] - m[d]);
    ex[t] = v;
    atomicAdd(&den[d], v);
}

// 64 lanes per edge -> coalesced feature gather from h[src]
__global__ void edge_scatter(const int* __restrict__ src, const int* __restrict__ dst,
                             int E, const float* __restrict__ ex,
                             const float* __restrict__ den,
                             const float* __restrict__ H, int F, int ldh,
                             float* __restrict__ out, int ldo)
{
    const long tid = (long)blockIdx.x * blockDim.x + threadIdx.x;
    const int eidx = (int)(tid >> 6);
    const int f    = (int)(tid & 63);
    if (eidx >= E || f >= F) return;
    const int d = dst[eidx], s = src[eidx];
    const float alpha = ex[eidx] / (den[d] + 1e-16f);
    atomicAdd(&out[(size_t)d * ldo + f], alpha * H[(size_t)s * ldh + f]);
}

__global__ void bias_add(float* __restrict__ X, int N, int ld, int F,
                         const float* __restrict__ b)
{
    const int t = blockIdx.x * blockDim.x + threadIdx.x;
    const int i = t / F, f = t - i * F;
    if (i >= N) return;
    X[(size_t)i * ld + f] += b[f];
}

// ---------------------------------------------------------------------------
// Host orchestration
// ---------------------------------------------------------------------------
extern "C" void kernel_launch(void* const* d_in, const int* in_sizes, int n_in,
                              void* d_out, int out_size, void* d_ws, size_t ws_size,
                              hipStream_t stream)
{
    (void)n_in; (void)out_size; (void)ws_size;

    const float* x    = (const float*)d_in[0];
    const int*   src  = (const int*)d_in[1];
    const int*   dst  = (const int*)d_in[2];
    const float* w1   = (const float*)d_in[3];  const float* b1  = (const float*)d_in[4];
    const float* w2   = (const float*)d_in[5];  const float* b2  = (const float*)d_in[6];
    const float* w3   = (const float*)d_in[7];  const float* b3  = (const float*)d_in[8];
    const float* cv1  = (const float*)d_in[9];  const float* g1g = (const float*)d_in[10];
    const float* g1bb = (const float*)d_in[11];
    const float* gw1  = (const float*)d_in[12]; const float* ga1 = (const float*)d_in[13];
    const float* gb1  = (const float*)d_in[14];
    const float* cv2  = (const float*)d_in[15]; const float* g2g = (const float*)d_in[16];
    const float* g2bb = (const float*)d_in[17];
    const float* gw2  = (const float*)d_in[18]; const float* ga2 = (const float*)d_in[19];
    const float* gb2  = (const float*)d_in[20];
    const float* cv3  = (const float*)d_in[21]; const float* g3g = (const float*)d_in[22];
    const float* g3bb = (const float*)d_in[23];

    const int N = in_sizes[0] / 64;
    const int E = in_sizes[1];
    float* outf = (float*)d_out;

    // workspace carve-up (floats)
    float* ws   = (float*)d_ws;
    float* buf0 = ws;                           // N*256
    float* buf1 = buf0 + (size_t)N * 256;       // N*256
    float* eS   = buf1 + (size_t)N * 256;       // E
    float* exS  = eS   + (size_t)E;             // E
    float* pd   = exS  + (size_t)E;             // N
    float* ps   = pd   + N;                     // N
    float* mS   = ps   + N;                     // N
    float* dS   = mS   + N;                     // N
    float* bnS  = dS   + N;                     // 192
    float* bnQ  = bnS  + 192;                   // 192
    float* wT   = bnQ  + 192;                   // 65536 (transposed-weight scratch)

    // GEMM with one-shot weight transpose into wT (stream-ordered, so wT reuse
    // across successive GEMMs is safe).
    auto gemm = [&](const float* A, int M, int K, int lda,
                    const float* W, int Nc, const float* bias,
                    float* C, int ldc, float slope) {
        transpose_w<<<(K * Nc + 255) / 256, 256, 0, stream>>>(W, K, Nc, wT);
        dim3 grid((Nc + 31) / 32, (M + 15) / 16);
        wmma_gemm_bias_act<<<grid, dim3(32), 0, stream>>>(
            A, M, K, lda, wT, Nc, bias, C, ldc, slope);
    };

    auto dense_block = [&](float* buf, int ld, int c0, const float* convW,
                           int pitch, const float* g, const float* bset) {
        int c = c0;
        for (int i = 0; i < 6; ++i) {
            bn_stats<<<c, 256, 0, stream>>>(buf, N, ld, bnS, bnQ);
            bn_act_conv_append<<<(N + 255) / 256, 256, 0, stream>>>(
                buf, N, ld, c, convW + (size_t)i * GROW * pitch, pitch,
                g + (size_t)i * pitch, bset + (size_t)i * pitch, bnS, bnQ, buf);
            c += GROW;
        }
    };

    auto gat = [&](const float* xin, int ldin, int K, const float* W, int F,
                   const float* avec, const float* bvec, float* h,
                   float* outbuf, int ldo) {
        gemm(xin, N, K, ldin, W, F, nullptr, h, F, 1.0f);          // h = x@W (linear)
        gat_node_proj<<<(N + 255) / 256, 256, 0, stream>>>(h, N, F, avec, pd, ps);
        fill_f32<<<(N + 255) / 256, 256, 0, stream>>>(mS, (size_t)N, -INFINITY);
        fill_f32<<<(N + 255) / 256, 256, 0, stream>>>(dS, (size_t)N, 0.f);
        edge_logit_max<<<(E + 255) / 256, 256, 0, stream>>>(src, dst, E, pd, ps, eS, mS);
        fix_m<<<(N + 255) / 256, 256, 0, stream>>>(mS, N);
        edge_exp_den<<<(E + 255) / 256, 256, 0, stream>>>(dst, E, eS, mS, exS, dS);
        size_t fillN = (size_t)N * ldo;
        fill_f32<<<(unsigned)((fillN + 255) / 256), 256, 0, stream>>>(outbuf, fillN, 0.f);
        long tot = (long)E * 64;
        edge_scatter<<<(unsigned)((tot + 255) / 256), 256, 0, stream>>>(
            src, dst, E, exS, dS, h, F, F, outbuf, ldo);
        bias_add<<<(unsigned)(((size_t)N * F + 255) / 256), 256, 0, stream>>>(
            outbuf, N, ldo, F, bvec);
    };

    // ---- MLP: 64 -> 256 -> 256 -> 24 (written into dense1 buffer, stride 96)
    gemm(x,    N, 64,  64,  w1, 256, b1, buf0, 256, 0.01f);
    gemm(buf0, N, 256, 256, w2, 256, b2, buf1, 256, 0.01f);
    gemm(buf1, N, 256, 256, w3, 24,  b3, buf0, 96,  0.01f);

    // ---- Dense block 1: 24 -> 96 (buf0, stride 96)
    dense_block(buf0, 96, 24, cv1, 84, g1g, g1bb);

    // ---- GAT 1: (N,96) -> (N,48), result placed in dense2 buffer (buf0, stride 120)
    gat(buf0, 96, 96, gw1, 48, ga1, gb1, buf1, buf0, 120);

    // ---- Dense block 2: 48 -> 120 (buf0, stride 120)
    dense_block(buf0, 120, 48, cv2, 108, g2g, g2bb);

    // ---- GAT 2: (N,120) -> (N,60), result placed in output buffer (stride 132)
    gat(buf0, 120, 120, gw2, 60, ga2, gb2, buf1, outf, 132);

    // ---- Dense block 3: 60 -> 132 (d_out, stride 132)
    dense_block(outf, 132, 60, cv3, 120, g3g, g3bb);
}